// RNNModelNoTeacherForcing_77790447665717
// MI455X (gfx1250) — compile-verified
//
#include <hip/hip_runtime.h>
#include <cmath>

// ---------------------------------------------------------------------------
// Autoregressive 2-layer tanh RNN (no teacher forcing), fp32, CDNA5 WMMA.
// V=32000, E=H=1024, L=2, B=64, S=256 (255 generation steps).
// All GEMMs run on V_WMMA_F32_16X16X4_F32 (fp32 matrix pipe, wave32).
// Weights (~139 MB) fit in the 192 MB L2 -> per-step weight traffic is
// L2-resident after the first step.
//
// fc projection (dominant GEMM, 2.1 GMAC/step): register-blocked 4 M-tiles
// per wave so the B (weight) fragment is loaded once and reused by 4 WMMAs
// (1.25 loads/WMMA instead of 2).
// RNN layers (serial-latency critical, 510 on the chain): 1 tile per wave,
// 256 waves, minimal latency.
// ---------------------------------------------------------------------------

typedef __attribute__((ext_vector_type(2))) float v2f;
typedef __attribute__((ext_vector_type(8))) float v8f;

#define VSZ   32000
#define ESZ   1024
#define HSZ   1024
#define BSZ   64
#define SEQ   256

// ---------------------------------------------------------------------------
// init: x0 = emb[tokens[:,0]]; h(layer0), h(layer1) = 0
// ---------------------------------------------------------------------------
__global__ void init_state(const int* __restrict__ xtok,
                           const float* __restrict__ emb,
                           float* __restrict__ xbuf,
                           float* __restrict__ hA0,
                           float* __restrict__ hB0) {
    const int b   = blockIdx.x;
    const int tid = threadIdx.x;
    const long tok = (long)xtok[b * SEQ];   // x[b, 0]
    for (int i = tid; i < ESZ; i += 256) {
        xbuf[b * ESZ + i] = emb[tok * ESZ + i];
        hA0[b * HSZ + i]  = 0.0f;
        hB0[b * HSZ + i]  = 0.0f;
    }
}

// ---------------------------------------------------------------------------
// One RNN layer: hout = tanh(xin @ Wih^T + b_ih + hprev @ Whh^T + b_hh)
// Grid: 64 blocks (N tiles of 16 over H=1024) x 128 threads = 4 waves.
// Wave w computes the 16x16 tile rows [16w,16w+16), cols [16*bx,16*bx+16).
//   A fragment (16x4):  lane(half=l>>4, m=l&15) holds A[m][k+2*half .. +1]
//   B fragment (4x16):  lane(half, n=l&15)      holds W[n][k+2*half .. +1]
//   C/D (16x16, v8f):   c[r] = out[r + 8*half][n]
// ---------------------------------------------------------------------------
__global__ void rnn_layer(const float* __restrict__ xin,
                          const float* __restrict__ hprev,
                          const float* __restrict__ wih,
                          const float* __restrict__ whh,
                          const float* __restrict__ bih,
                          const float* __restrict__ bhh,
                          float* __restrict__ hout) {
    const int lane = threadIdx.x & 31;
    const int wave = threadIdx.x >> 5;       // 0..3 -> M tile
    const int m0   = wave * 16;
    const int n0   = blockIdx.x * 16;
    const int half = lane >> 4;
    const int mn   = lane & 15;

    const float* arow0 = xin   + (long)(m0 + mn) * ESZ + 2 * half;   // A: x
    const float* brow0 = wih   + (long)(n0 + mn) * ESZ + 2 * half;   // B: Wih
    const float* arow1 = hprev + (long)(m0 + mn) * HSZ + 2 * half;   // A: h
    const float* brow1 = whh   + (long)(n0 + mn) * HSZ + 2 * half;   // B: Whh

    __builtin_prefetch(brow0, 0, 3);
    __builtin_prefetch(brow1, 0, 3);

    v8f acc = {};
#pragma unroll 8
    for (int k = 0; k < ESZ; k += 4) {
        v2f a = *(const v2f*)(arow0 + k);
        v2f b = *(const v2f*)(brow0 + k);
        acc = __builtin_amdgcn_wmma_f32_16x16x4_f32(
            false, a, false, b, (short)0, acc, false, false);
    }
#pragma unroll 8
    for (int k = 0; k < HSZ; k += 4) {
        v2f a = *(const v2f*)(arow1 + k);
        v2f b = *(const v2f*)(brow1 + k);
        acc = __builtin_amdgcn_wmma_f32_16x16x4_f32(
            false, a, false, b, (short)0, acc, false, false);
    }

    const float bias = bih[n0 + mn] + bhh[n0 + mn];
#pragma unroll
    for (int r = 0; r < 8; ++r) {
        const int m = m0 + r + 8 * half;
        hout[(long)m * HSZ + n0 + mn] = tanhf(acc[r] + bias);
    }
}

// ---------------------------------------------------------------------------
// Output projection: logits = h2 @ fc_w^T + fc_b   (64 x 32000)
// Register-blocked: each wave owns ONE 16-wide N-tile and ALL FOUR M-tiles
// (B=64). Per k-step: 1 B-fragment load, reused by 4 WMMAs into 4 accs.
// Grid: 500 blocks x 128 threads (4 waves = 4 N-tiles per block).
// ---------------------------------------------------------------------------
__global__ void fc_logits(const float* __restrict__ h2,
                          const float* __restrict__ fcw,
                          const float* __restrict__ fcb,
                          float* __restrict__ logits) {
    const int lane = threadIdx.x & 31;
    const int wave = threadIdx.x >> 5;                 // 0..3 -> N sub-tile
    const int n0   = (blockIdx.x * 4 + wave) * 16;
    const int half = lane >> 4;
    const int mn   = lane & 15;

    const float* brow = fcw + (long)(n0 + mn) * HSZ + 2 * half;
    const float* a0   = h2  + (long)(mn)      * HSZ + 2 * half;   // M-tile 0
    const float* a1   = a0 + 16L * HSZ;                            // M-tile 1
    const float* a2   = a0 + 32L * HSZ;                            // M-tile 2
    const float* a3   = a0 + 48L * HSZ;                            // M-tile 3

    __builtin_prefetch(brow, 0, 3);

    v8f c0 = {}, c1 = {}, c2 = {}, c3 = {};
#pragma unroll 4
    for (int k = 0; k < HSZ; k += 4) {
        v2f b = *(const v2f*)(brow + k);               // shared weight frag
        v2f a;
        a = *(const v2f*)(a0 + k);
        c0 = __builtin_amdgcn_wmma_f32_16x16x4_f32(false, a, false, b,
                                                   (short)0, c0, false, false);
        a = *(const v2f*)(a1 + k);
        c1 = __builtin_amdgcn_wmma_f32_16x16x4_f32(false, a, false, b,
                                                   (short)0, c1, false, false);
        a = *(const v2f*)(a2 + k);
        c2 = __builtin_amdgcn_wmma_f32_16x16x4_f32(false, a, false, b,
                                                   (short)0, c2, false, false);
        a = *(const v2f*)(a3 + k);
        c3 = __builtin_amdgcn_wmma_f32_16x16x4_f32(false, a, false, b,
                                                   (short)0, c3, false, false);
    }

    const float bias = fcb[n0 + mn];
#pragma unroll
    for (int r = 0; r < 8; ++r) {
        const int m = r + 8 * half;
        logits[(long)(m)      * VSZ + n0 + mn] = c0[r] + bias;
        logits[(long)(m + 16) * VSZ + n0 + mn] = c1[r] + bias;
        logits[(long)(m + 32) * VSZ + n0 + mn] = c2[r] + bias;
        logits[(long)(m + 48) * VSZ + n0 + mn] = c3[r] + bias;
    }
}

// ---------------------------------------------------------------------------
// Per-row argmax over 32000 logits (first-max tie-break like jnp.argmax),
// then gather next input: x[b] = emb[pred].  Grid: 64 blocks x 256 threads.
// ---------------------------------------------------------------------------
__global__ void argmax_embed(const float* __restrict__ logits,
                             const float* __restrict__ emb,
                             float* __restrict__ xout) {
    __shared__ float smax[256];
    __shared__ int   sidx[256];
    const int b   = blockIdx.x;
    const int tid = threadIdx.x;

    float best = -__builtin_huge_valf();
    int   bidx = 0;
    for (int i = tid; i < VSZ; i += 256) {
        const float v = logits[(long)b * VSZ + i];
        if (v > best) { best = v; bidx = i; }
    }
    smax[tid] = best;
    sidx[tid] = bidx;
    __syncthreads();
    for (int s = 128; s > 0; s >>= 1) {
        if (tid < s) {
            const float vo = smax[tid + s];
            const int   io = sidx[tid + s];
            if (vo > smax[tid] || (vo == smax[tid] && io < sidx[tid])) {
                smax[tid] = vo; sidx[tid] = io;
            }
        }
        __syncthreads();
    }
    const long pred = (long)sidx[0];
    for (int i = tid; i < ESZ; i += 256)
        xout[b * ESZ + i] = emb[pred * ESZ + i];
}

// ---------------------------------------------------------------------------
// Host-side orchestration: 255 sequential steps, ping-pong hidden buffers.
// ---------------------------------------------------------------------------
extern "C" void kernel_launch(void* const* d_in, const int* in_sizes, int n_in,
                              void* d_out, int out_size, void* d_ws, size_t ws_size,
                              hipStream_t stream) {
    const int*   xtok = (const int*)  d_in[0];   // (64, 256)
    const float* emb  = (const float*)d_in[1];   // (32000, 1024)
    const float* wih  = (const float*)d_in[2];   // (2, 1024, 1024)
    const float* whh  = (const float*)d_in[3];   // (2, 1024, 1024)
    const float* bih  = (const float*)d_in[4];   // (2, 1024)
    const float* bhh  = (const float*)d_in[5];   // (2, 1024)
    const float* fcw  = (const float*)d_in[6];   // (32000, 1024)
    const float* fcb  = (const float*)d_in[7];   // (32000,)
    float* logits = (float*)d_out;               // (64, 32000)

    // Workspace: x(64K) + 4 hidden buffers(64K each) = 1.25 MB fp32
    float* xbuf  = (float*)d_ws;
    float* hA[2] = { xbuf + BSZ * ESZ,     xbuf + 2 * BSZ * ESZ };
    float* hB[2] = { xbuf + 3 * BSZ * ESZ, xbuf + 4 * BSZ * ESZ };

    init_state<<<BSZ, 256, 0, stream>>>(xtok, emb, xbuf, hA[0], hB[0]);

    for (int t = 0; t < SEQ - 1; ++t) {
        const int p = t & 1;
        // layer 0: x -> hA
        rnn_layer<<<HSZ / 16, 128, 0, stream>>>(
            xbuf, hA[p],
            wih, whh, bih, bhh,
            hA[1 - p]);
        // layer 1: hA -> hB
        rnn_layer<<<HSZ / 16, 128, 0, stream>>>(
            hA[1 - p], hB[p],
            wih + (long)HSZ * ESZ, whh + (long)HSZ * HSZ,
            bih + HSZ, bhh + HSZ,
            hB[1 - p]);
        // logits = hB @ fc_w^T + fc_b  (kept in d_out; final step's survives)
        fc_logits<<<VSZ / 64, 128, 0, stream>>>(hB[1 - p], fcw, fcb, logits);
        // next input token embedding
        argmax_embed<<<BSZ, 256, 0, stream>>>(logits, emb, xbuf);
    }
}